// StableVMD_45174466019559
// MI455X (gfx1250) — compile-verified
//
#include <hip/hip_runtime.h>
#include <hip/hip_bf16.h>

// ============================================================================
// StableVMD on MI455X (gfx1250, wave32).
//
//   K1: TDM-load row -> LDS, normalize + rfft(8192)  -> x_hat     (128 blocks)
//   K2: 50 Gauss-Seidel VMD sweeps, ALL state in LDS -> u_hat     (128 blocks)
//       (u_hat[5][4104] re+im + running sum + x_hat ~= 230 KB, fits the
//        320 KB CDNA5 WGP LDS -> zero global traffic inside the solver loop)
//   K3: per-(row,mode) irfft(8192) + denorm          -> out       (640 blocks)
//
// FFT-8192 = four-step: N = 512 * 16, n = 16*n1 + n2, k = k1 + 512*k2.
//   (a) 16 radix-2 DIT FFT-512s (rows n2) in LDS, bit-reversed input
//   (b) fused twiddle  Z[n2][k1] = Y[n2][k1] * e^{-2*pi*i*n2*k1/8192}
//   (c) DFT-16:  X[16 x 512] = W16[16 x 16] @ Z[16 x 512]  (complex) via
//       V_WMMA_F32_16X16X4_F32 (full-precision f32 matrix pipe).
//
// TAU = 0 => lam == 0 identically (folded out).  The global `done` freeze
// only freezes state after convergence; we run all 50 sweeps instead of a
// cross-grid reduction (differs from frozen state by < TOL).
// ============================================================================

typedef float v2f __attribute__((ext_vector_type(2)));
typedef float v8f __attribute__((ext_vector_type(8)));
typedef unsigned int v4u __attribute__((ext_vector_type(4)));
typedef int v8i __attribute__((ext_vector_type(8)));
typedef int v4i __attribute__((ext_vector_type(4)));

#define NB   128
#define NT   8192
#define NK   5
#define FB   4097
#define FBP  4104            // padded row stride for spectra
#define NITER 50
#define ALPHA_C 2000.0f

__device__ __forceinline__ int brev9(int x) { return (int)(__brev((unsigned)x) >> 23); }

// ---------------------------------------------------------------------------
// (a) 9 radix-2 DIT stages on each of 16 rows of 512 (bit-reversed input).
// ---------------------------------------------------------------------------
__device__ __forceinline__ void fft_stages(float* lre, float* lim, int tid) {
  for (int s = 1; s <= 9; ++s) {
    __syncthreads();
    const int half = 1 << (s - 1);
    const int pos  = tid & (half - 1);
    const int k0   = ((tid >> (s - 1)) << s) + pos;
    float wr, wi;
    __sincosf(-3.14159265358979323846f * (float)pos / (float)half, &wi, &wr);
    #pragma unroll 4
    for (int row = 0; row < 16; ++row) {
      const int base = row << 9;
      float ar = lre[base + k0],        ai = lim[base + k0];
      float br = lre[base + k0 + half], bi = lim[base + k0 + half];
      float tr = br * wr - bi * wi;
      float ti = br * wi + bi * wr;
      lre[base + k0]        = ar + tr;  lim[base + k0]        = ai + ti;
      lre[base + k0 + half] = ar - tr;  lim[base + k0 + half] = ai - ti;
    }
  }
  __syncthreads();
}

// ---------------------------------------------------------------------------
// (b)+(c) fused twiddle + DFT-16 via V_WMMA_F32_16X16X4_F32, in place.
// A (16x4 f32, ISA 7.12.2): lanes 0-15 -> {K0,K1}, lanes 16-31 -> {K2,K3}.
// B (4x16  f32): VGPR0 -> rows {0 | 2}, VGPR1 -> rows {1 | 3} (C/D analogue).
// C/D (16x16 f32): VGPR j -> row j (lanes 0-15) / row j+8 (lanes 16-31).
// Each wave owns 4 disjoint 16-column tiles -> in-place overwrite is safe.
// ---------------------------------------------------------------------------
__device__ __forceinline__ void fft_wmma_stage(float* lre, float* lim, int tid) {
  const float TWO_PI_16 = 0.39269908169872414f;      // 2*pi/16
  const float TWO_PI_N  = 7.6699039394282217e-4f;    // 2*pi/8192
  const int lane = tid & 31;
  const int wave = tid >> 5;
  const int mrow = lane & 15;
  const int hi   = lane >> 4;

  v2f ac[4], asn[4], ans[4];
  #pragma unroll
  for (int c = 0; c < 4; ++c) {
    const int kk0 = 4 * c + 2 * hi;
    float s0, c0, s1, c1;
    __sincosf(TWO_PI_16 * (float)((mrow * kk0) & 15), &s0, &c0);
    __sincosf(TWO_PI_16 * (float)((mrow * (kk0 + 1)) & 15), &s1, &c1);
    ac[c].x  = c0;  ac[c].y  = c1;
    asn[c].x = s0;  asn[c].y = s1;
    ans[c].x = -s0; ans[c].y = -s1;
  }

  #pragma unroll
  for (int t = 0; t < 4; ++t) {
    const int col = (wave * 4 + t) * 16 + mrow;      // k1 in [0, 512)
    v2f zr[4], zi[4];
    #pragma unroll
    for (int c = 0; c < 4; ++c) {
      const int r0 = 4 * c + 2 * hi;
      float yr0 = lre[(r0 << 9) + col],       yi0 = lim[(r0 << 9) + col];
      float yr1 = lre[((r0 + 1) << 9) + col], yi1 = lim[((r0 + 1) << 9) + col];
      float s0, c0, s1, c1;
      __sincosf(-TWO_PI_N * (float)((r0 * col) & 8191), &s0, &c0);
      __sincosf(-TWO_PI_N * (float)(((r0 + 1) * col) & 8191), &s1, &c1);
      zr[c].x = yr0 * c0 - yi0 * s0;  zi[c].x = yr0 * s0 + yi0 * c0;
      zr[c].y = yr1 * c1 - yi1 * s1;  zi[c].y = yr1 * s1 + yi1 * c1;
    }
    v8f aR = {0.f,0.f,0.f,0.f,0.f,0.f,0.f,0.f};
    v8f aI = {0.f,0.f,0.f,0.f,0.f,0.f,0.f,0.f};
    #pragma unroll
    for (int c = 0; c < 4; ++c) {
      // Re = Wr@Zr + sin@Zi ; Im = Wr@Zi + (-sin)@Zr   (K accumulated 4-wide)
      aR = __builtin_amdgcn_wmma_f32_16x16x4_f32(false, ac[c],  false, zr[c], (short)0, aR, false, false);
      aR = __builtin_amdgcn_wmma_f32_16x16x4_f32(false, asn[c], false, zi[c], (short)0, aR, false, false);
      aI = __builtin_amdgcn_wmma_f32_16x16x4_f32(false, ac[c],  false, zi[c], (short)0, aI, false, false);
      aI = __builtin_amdgcn_wmma_f32_16x16x4_f32(false, ans[c], false, zr[c], (short)0, aI, false, false);
    }
    #pragma unroll
    for (int j = 0; j < 8; ++j) {
      const int m = j + 8 * hi;                      // k2 row
      lre[(m << 9) + col] = aR[j];                   // flat idx == 512*k2+k1 == bin k
      lim[(m << 9) + col] = aI[j];
    }
  }
  __syncthreads();
}

__device__ __forceinline__ void fft8192_core(float* lre, float* lim, int tid) {
  fft_stages(lre, lim, tid);
  fft_wmma_stage(lre, lim, tid);
}

// ---------------------------------------------------------------------------
// TDM: DMA one contiguous row of `words` dwords from global into LDS.
// D# per ISA 8.3/8.4: group0 = {count=1, lds_addr, global_addr, type=2};
// group1 = 1-D tile (data_size=4B, tensor_dim0=tile_dim0=words, dim1=1).
// 6-arg clang-23 builtin form: (v4u g0, v8i g1, v4i g2, v4i g3, v8i g4, cpol).
// Issued by the calling wave only; waits on tensorcnt itself; caller must barrier.
// ---------------------------------------------------------------------------
__device__ __forceinline__ void tdm_load_row(const float* gsrc, float* ldst,
                                             unsigned words) {
  unsigned long long ga = (unsigned long long)(uintptr_t)gsrc;
  unsigned lo = (unsigned)ga;
  unsigned hi = (unsigned)(ga >> 32);
  unsigned ldsoff = (unsigned)(uintptr_t)(void*)ldst;   // low 32 bits = LDS byte addr
  v4u g0;
  g0.x = 1u;                                     // count=1, is_restore=0, no gather
  g0.y = ldsoff;                                 // lds_addr
  g0.z = lo;                                     // global_addr[31:0]
  g0.w = (hi & 0x01FFFFFFu) | (2u << 30);        // global_addr[56:32] | type=2
  v8i g1;
  g1[0] = (int)(2u << 16);                       // wg_mask=0, data_size=2 (4 bytes)
  g1[1] = (int)((words & 0xFFFFu) << 16);        // tensor_dim0[15:0]
  g1[2] = (int)(((words >> 16) & 0xFFFFu) | (1u << 16)); // t_dim0[31:16] | t_dim1=1
  g1[3] = (int)((words & 0xFFFFu) << 16);        // tile_dim0  (dim1 hi16 = 0)
  g1[4] = 1;                                     // tile_dim1=1, tile_dim2=0
  g1[5] = (int)words;                            // tensor_dim0_stride[31:0]
  g1[6] = 0;                                     // stride0 hi | stride1 lo
  g1[7] = 0;
  v4i gz4 = {0, 0, 0, 0};
  v8i gz8 = {0, 0, 0, 0, 0, 0, 0, 0};
  __builtin_amdgcn_tensor_load_to_lds(g0, g1, gz4, gz4, gz8, 0);
  __builtin_amdgcn_s_wait_tensorcnt(0);
}

// ============================ Kernel 1: normalize + rfft =====================
__global__ __launch_bounds__(256)
void k_norm_fft(const float* __restrict__ sig, float* __restrict__ xr,
                float* __restrict__ xi, float* __restrict__ ms) {
  __shared__ float lre[NT];
  __shared__ float lim[NT];
  float* red = lre;                       // overlay: lre unused until scatter
  const int b = blockIdx.x, tid = threadIdx.x;

  if (tid < 32)                           // wave 0 issues the DMA
    tdm_load_row(sig + (size_t)b * NT, lim, NT);
  __syncthreads();

  float s1 = 0.f, s2 = 0.f;
  for (int n = tid; n < NT; n += 256) {
    float v = lim[n];
    s1 += v; s2 += v * v;
  }
  red[tid] = s1; __syncthreads();
  for (int off = 128; off > 0; off >>= 1) {
    if (tid < off) red[tid] += red[tid + off];
    __syncthreads();
  }
  const float mean = red[0] * (1.0f / NT);
  __syncthreads();
  red[tid] = s2; __syncthreads();
  for (int off = 128; off > 0; off >>= 1) {
    if (tid < off) red[tid] += red[tid + off];
    __syncthreads();
  }
  const float var  = (red[0] - (float)NT * mean * mean) * (1.0f / (NT - 1));
  const float stdv = sqrtf(fmaxf(var, 0.f)) + 1e-8f;
  if (tid == 0) { ms[2 * b] = mean; ms[2 * b + 1] = stdv; }
  __syncthreads();                        // reads of red done before scatter

  const float inv = 1.0f / stdv;
  for (int n = tid; n < NT; n += 256) {   // x[16*n1+n2] -> row n2, col brev9(n1)
    float xn = (lim[n] - mean) * inv;
    lre[((n & 15) << 9) + brev9(n >> 4)] = xn;
  }
  __syncthreads();
  for (int n = tid; n < NT; n += 256) lim[n] = 0.f;

  fft8192_core(lre, lim, tid);

  for (int f = tid; f < FB; f += 256) {
    xr[(size_t)b * FBP + f] = lre[f];
    xi[(size_t)b * FBP + f] = lim[f];
  }
}

// ===================== Kernel 2: 50 Gauss-Seidel VMD sweeps ==================
// Entire per-row solver state lives in LDS (~230 KB of the 320 KB WGP LDS):
// no global traffic inside the 50 x 5 sweep loop.
__global__ __launch_bounds__(256)
void k_vmd(const float* __restrict__ xr_g, const float* __restrict__ xi_g,
           float* __restrict__ ur_g, float* __restrict__ ui_g) {
  __shared__ float url[NK * FBP];         // 82,080 B
  __shared__ float uil[NK * FBP];         // 82,080 B
  __shared__ float sr[FBP], si[FBP];      // 32,832 B
  __shared__ float xr[FBP], xi[FBP];      // 32,832 B
  __shared__ float r0[256], r1[256];      //  2,048 B
  const int b = blockIdx.x, tid = threadIdx.x;
  const float* xrb = xr_g + (size_t)b * FBP;
  const float* xib = xi_g + (size_t)b * FBP;

  for (int f = tid; f < FBP; f += 256) {
    sr[f] = 0.f; si[f] = 0.f;
    xr[f] = xrb[f]; xi[f] = xib[f];
  }
  for (int v = tid; v < NK * FBP; v += 256) { url[v] = 0.f; uil[v] = 0.f; }
  float om[NK] = {0.f, 0.f, 0.f, 0.f, 0.f};
  __syncthreads();

  for (int it = 0; it < NITER; ++it) {
    float omn[NK];
    for (int k = 0; k < NK; ++k) {        // Gauss-Seidel over modes (lam == 0)
      const float omk = om[k];
      float a0 = 0.f, a1 = 0.f;
      for (int f = tid; f < FB; f += 256) {   // thread exclusively owns bin f
        float uro = url[k * FBP + f], uio = uil[k * FBP + f];
        float nr = xr[f] - (sr[f] - uro);
        float ni = xi[f] - (si[f] - uio);
        float fr = (float)f * (1.0f / NT);
        float dd = fr - omk;
        float den = 1.0f / (1.0f + ALPHA_C * dd * dd);
        nr *= den; ni *= den;
        sr[f] += nr - uro; si[f] += ni - uio;
        url[k * FBP + f] = nr; uil[k * FBP + f] = ni;
        float p = nr * nr + ni * ni;
        a0 += p; a1 += p * fr;
      }
      r0[tid] = a0; r1[tid] = a1; __syncthreads();
      for (int off = 128; off > 0; off >>= 1) {
        if (tid < off) { r0[tid] += r0[tid + off]; r1[tid] += r1[tid + off]; }
        __syncthreads();
      }
      omn[k] = r1[0] / (r0[0] + 1e-10f);  // applied after the full sweep
      __syncthreads();
    }
    for (int k = 0; k < NK; ++k) om[k] = omn[k];
  }

  float* urb = ur_g + (size_t)b * NK * FBP;
  float* uib = ui_g + (size_t)b * NK * FBP;
  for (int v = tid; v < NK * FBP; v += 256) { urb[v] = url[v]; uib[v] = uil[v]; }
}

// ================= Kernel 3: irfft per (row, mode) + denorm ==================
// irfft(U) = Re[ FFT( conj(Ufull) ) ] / N ; Ufull Hermitian-extended.
__global__ __launch_bounds__(256)
void k_irfft(const float* __restrict__ ur, const float* __restrict__ ui,
             const float* __restrict__ ms, float* __restrict__ out) {
  __shared__ float lre[NT];
  __shared__ float lim[NT];
  const int bk = blockIdx.x, tid = threadIdx.x;
  const int b = bk / NK;
  const float* urb = ur + (size_t)bk * FBP;
  const float* uib = ui + (size_t)bk * FBP;

  for (int n = tid; n < NT; n += 256) {
    float r, m;
    if (n <= NT / 2) { r = urb[n];      m = -uib[n]; }       // conj(U[n])
    else             { r = urb[NT - n]; m =  uib[NT - n]; }  // conj(conj(U)) = U
    const int d = ((n & 15) << 9) + brev9(n >> 4);
    lre[d] = r; lim[d] = m;
  }

  fft8192_core(lre, lim, tid);

  const float mean = ms[2 * b], stdv = ms[2 * b + 1];
  const float sc = stdv * (1.0f / NT);
  const float av = mean * (1.0f / NK);
  float* ob = out + (size_t)bk * NT;
  for (int t = tid; t < NT; t += 256) ob[t] = lre[t] * sc + av;
}

// ============================================================================
extern "C" void kernel_launch(void* const* d_in, const int* in_sizes, int n_in,
                              void* d_out, int out_size, void* d_ws, size_t ws_size,
                              hipStream_t stream) {
  (void)in_sizes; (void)n_in; (void)out_size; (void)ws_size;
  const float* sig = (const float*)d_in[0];
  float* ws = (float*)d_ws;
  float* xr = ws;                                  // [128][4104]
  float* xi = xr + (size_t)NB * FBP;               // [128][4104]
  float* ur = xi + (size_t)NB * FBP;               // [128][5][4104]
  float* ui = ur + (size_t)NB * NK * FBP;          // [128][5][4104]
  float* ms = ui + (size_t)NB * NK * FBP;          // [128][2] mean,std

  k_norm_fft<<<NB, 256, 0, stream>>>(sig, xr, xi, ms);
  k_vmd     <<<NB, 256, 0, stream>>>(xr, xi, ur, ui);
  k_irfft   <<<NB * NK, 256, 0, stream>>>(ur, ui, ms, (float*)d_out);
}